// V2FConv3d_54640573939774
// MI455X (gfx1250) — compile-verified
//
#include <hip/hip_runtime.h>
#include <hip/hip_bf16.h>

typedef float v2f_t __attribute__((ext_vector_type(2)));
typedef float v8f_t __attribute__((ext_vector_type(8)));

#define GRID_MAIN 1024
#define BN_EPS_F 1e-3f

// ---------------------------------------------------------------------------
// Kernel 0: spatial_filter[3][64] = spharm(3x9, hardcoded) @ spatial_weights(9x64)
// spharm rows are Y_{l,m} at the unit axes e_x, e_y, e_z (L=2, Condon-Shortley
// phase per the reference recurrence). sin(pi)-type terms (~1e-16) are 0 in f32.
// ---------------------------------------------------------------------------
__global__ __launch_bounds__(64) void v2f_filt_kernel(const float* __restrict__ sw,
                                                      float* __restrict__ filt) {
  const float A  = 0.28209479177387814f;  // 1/2 sqrt(1/pi)
  const float B  = 0.4886025119029199f;   // sqrt(3/(4pi))
  const float Cc = 0.31539156525252005f;  // 1/2 sqrt(5/(4pi))
  const float D  = 0.5462742152960396f;   // 1/4 sqrt(15/pi)
  const float S[3][9] = {
      {A, 0.f, 0.f,  -B, 0.f, 0.f,     -Cc, 0.f,   D},   // e_x
      {A,  -B, 0.f, 0.f, 0.f, 0.f,     -Cc, 0.f,  -D},   // e_y
      {A, 0.f,   B, 0.f, 0.f, 0.f, 2.f*Cc, 0.f, 0.f}};   // e_z
  const int c = threadIdx.x;  // 0..63
  #pragma unroll
  for (int k = 0; k < 3; ++k) {
    float acc = 0.f;
    #pragma unroll
    for (int j = 0; j < 9; ++j) acc += S[k][j] * sw[j * 64 + c];
    filt[k * 64 + c] = acc;
  }
}

// ---------------------------------------------------------------------------
// Kernel 1: gather + depthwise 3-tap + [64x64]x[64x64] WMMA GEMM + bias + ReLU,
// writes pre-BN activations to out, and deterministic per-block BN partials.
// ---------------------------------------------------------------------------
__global__ __launch_bounds__(256) void v2f_main_kernel(
    const float* __restrict__ inputs, const long long* __restrict__ face,
    const float* __restrict__ dw, const float* __restrict__ bias,
    const float* __restrict__ filt, float* __restrict__ out,
    float* __restrict__ partS, float* __restrict__ partQ, int numTiles) {
  // stride 68 (= 4 mod 64 banks) makes both A- and B-fragment b64 loads
  // conflict-free: lower half-wave hits banks {4i+c,4i+c+1}, upper {4i+c+2,+3}.
  __shared__ __align__(16) float sWt[64][68];  // depth_weights TRANSPOSED [n][k]
  __shared__ __align__(16) float sV[64][68];   // v2f tile [face][k]
  __shared__ long long sIdx[192];              // 64 faces x 3 vertex ids
  __shared__ float sPs[8][64];                 // per-(wave-half) channel sums
  __shared__ float sPq[8][64];
  __shared__ float runS[64];                   // per-block running BN sums
  __shared__ float runQ[64];

  const int tid = threadIdx.x;

  // stage depth weights transposed: sWt[n][k] = dw[k][n]
  // -> each lane's B fragment (W[kk][col], W[kk+1][col]) is a contiguous b64
  for (int i4 = tid; i4 < 1024; i4 += 256) {
    const int k = i4 >> 4, n4 = (i4 & 15) << 2;
    const float4 wv = *(const float4*)(dw + k * 64 + n4);
    sWt[n4 + 0][k] = wv.x;
    sWt[n4 + 1][k] = wv.y;
    sWt[n4 + 2][k] = wv.z;
    sWt[n4 + 3][k] = wv.w;
  }
  if (tid < 64) { runS[tid] = 0.f; runQ[tid] = 0.f; }

  // per-thread spatial filter taps for its 4 channels (phase-1 mapping)
  const int c4 = (tid & 15) << 2;
  const float4 F0 = *(const float4*)(filt + 0 * 64 + c4);
  const float4 F1 = *(const float4*)(filt + 1 * 64 + c4);
  const float4 F2 = *(const float4*)(filt + 2 * 64 + c4);

  // WMMA wave geometry: wave w -> rows m..m+15, channels n0..n0+31
  const int lane = tid & 31;
  const int w    = tid >> 5;        // 0..7
  const int mt   = w & 3;
  const int m    = mt << 4;
  const int n0   = (w >> 2) << 5;
  const int hl   = lane >> 4;       // half-wave: selects K pair / row offset 8
  const int l16  = lane & 15;
  const int row  = m + l16;         // A: M = lane%16
  const int col0 = n0 + l16;        // B/D: N = lane%16
  const int col1 = col0 + 16;
  const float b0s = bias[col0];
  const float b1s = bias[col1];
  const int fr = tid >> 4;          // phase-1 face slot 0..15

  __syncthreads();

  for (int t = blockIdx.x; t < numTiles; t += gridDim.x) {
    const int fBase = t << 6;
    if (tid < 192) sIdx[tid] = face[(size_t)fBase * 3 + tid];
    __syncthreads();

    // phase 1: v2f[f][c] = sum_k inputs[face[f][k]][c] * filt[k][c]
    #pragma unroll
    for (int it = 0; it < 4; ++it) {
      const int f = (it << 4) + fr;
      const long long i0 = sIdx[f * 3 + 0];
      const long long i1 = sIdx[f * 3 + 1];
      const long long i2 = sIdx[f * 3 + 2];
      const float4 x0 = *(const float4*)(inputs + i0 * 64 + c4);
      const float4 x1 = *(const float4*)(inputs + i1 * 64 + c4);
      const float4 x2 = *(const float4*)(inputs + i2 * 64 + c4);
      float4 v;
      v.x = x0.x * F0.x + x1.x * F1.x + x2.x * F2.x;
      v.y = x0.y * F0.y + x1.y * F1.y + x2.y * F2.y;
      v.z = x0.z * F0.z + x1.z * F1.z + x2.z * F2.z;
      v.w = x0.w * F0.w + x1.w * F1.w + x2.w * F2.w;
      *(float4*)(&sV[f][c4]) = v;
    }
    __syncthreads();

    // phase 2: D = v2f_tile(16x64) @ W(64x32) per wave via f32 WMMA, K steps of 4
    v8f_t acc0 = {};
    v8f_t acc1 = {};
    #pragma unroll
    for (int k = 0; k < 64; k += 4) {
      const int kk = k + (hl << 1);   // lanes 0-15: K=k,k+1; lanes 16-31: K=k+2,k+3
      const v2f_t a  = *(const v2f_t*)(&sV[row][kk]);
      const v2f_t b0 = *(const v2f_t*)(&sWt[col0][kk]);
      const v2f_t b1 = *(const v2f_t*)(&sWt[col1][kk]);
      acc0 = __builtin_amdgcn_wmma_f32_16x16x4_f32(false, a, false, b0, (short)0,
                                                   acc0, false, false);
      acc1 = __builtin_amdgcn_wmma_f32_16x16x4_f32(false, a, false, b1, (short)0,
                                                   acc1, false, false);
    }

    // phase 3: bias + ReLU + store + per-channel BN partials
    // D layout: VGPR r -> row m + r + 8*hl, channel col0/col1 (fixed per lane)
    float s0 = 0.f, q0 = 0.f, s1 = 0.f, q1 = 0.f;
    const int rbase = fBase + m + (hl << 3);
    #pragma unroll
    for (int r = 0; r < 8; ++r) {
      float v0 = acc0[r] + b0s; v0 = fmaxf(v0, 0.f);
      float v1 = acc1[r] + b1s; v1 = fmaxf(v1, 0.f);
      const size_t ro = (size_t)(rbase + r) << 6;
      out[ro + col0] = v0;
      out[ro + col1] = v1;
      s0 += v0; q0 += v0 * v0;
      s1 += v1; q1 += v1 * v1;
    }
    const int h = (mt << 1) + hl;     // unique (h, channel) slot per lane
    sPs[h][col0] = s0; sPq[h][col0] = q0;
    sPs[h][col1] = s1; sPq[h][col1] = q1;
    __syncthreads();
    if (tid < 64) {                   // fixed-order tree: bitwise deterministic
      float as = 0.f, aq = 0.f;
      #pragma unroll
      for (int j = 0; j < 8; ++j) { as += sPs[j][tid]; aq += sPq[j][tid]; }
      runS[tid] += as; runQ[tid] += aq;
    }
    __syncthreads();
  }

  if (tid < 64) {
    partS[(size_t)blockIdx.x * 64 + tid] = runS[tid];
    partQ[(size_t)blockIdx.x * 64 + tid] = runQ[tid];
  }
}

// ---------------------------------------------------------------------------
// Kernel 2: reduce per-block partials -> per-channel scale/shift for BN
// ---------------------------------------------------------------------------
__global__ __launch_bounds__(64) void v2f_stats_kernel(
    const float* __restrict__ partS, const float* __restrict__ partQ,
    const float* __restrict__ gamma, const float* __restrict__ beta,
    float* __restrict__ params, int nPart, float invF) {
  const int c = threadIdx.x;
  float s = 0.f, q = 0.f;
  for (int b = 0; b < nPart; ++b) {
    s += partS[b * 64 + c];
    q += partQ[b * 64 + c];
  }
  const float mean = s * invF;
  const float var  = fmaxf(q * invF - mean * mean, 0.f);  // biased variance
  const float rstd = rsqrtf(var + BN_EPS_F);
  const float sc   = gamma[c] * rstd;
  params[c]      = sc;
  params[64 + c] = beta[c] - mean * sc;
}

// ---------------------------------------------------------------------------
// Kernel 3: out = out*scale[ch] + shift[ch]  (float4 grid-stride)
// ---------------------------------------------------------------------------
__global__ __launch_bounds__(256) void v2f_norm_kernel(float* __restrict__ out,
                                                       const float* __restrict__ params,
                                                       int n4) {
  __shared__ float sc[64];
  __shared__ float sh[64];
  const int tid = threadIdx.x;
  if (tid < 64) { sc[tid] = params[tid]; sh[tid] = params[64 + tid]; }
  __syncthreads();
  float4* o4 = (float4*)out;
  for (int i = blockIdx.x * 256 + tid; i < n4; i += gridDim.x * 256) {
    float4 v = o4[i];
    const int ch = (i << 2) & 63;   // 64%4==0 -> 4 contiguous channels
    v.x = v.x * sc[ch]     + sh[ch];
    v.y = v.y * sc[ch + 1] + sh[ch + 1];
    v.z = v.z * sc[ch + 2] + sh[ch + 2];
    v.w = v.w * sc[ch + 3] + sh[ch + 3];
    o4[i] = v;
  }
}

// ---------------------------------------------------------------------------
extern "C" void kernel_launch(void* const* d_in, const int* in_sizes, int n_in,
                              void* d_out, int out_size, void* d_ws, size_t ws_size,
                              hipStream_t stream) {
  const float*     inputs = (const float*)d_in[0];
  const long long* face   = (const long long*)d_in[1];   // int64 indices
  const float*     sw     = (const float*)d_in[2];       // [9][64]
  const float*     dw     = (const float*)d_in[3];       // [64][64]
  const float*     bias   = (const float*)d_in[4];       // [1][64]
  const float*     gamma  = (const float*)d_in[5];       // [64]
  const float*     beta   = (const float*)d_in[6];       // [64]
  float*           out    = (float*)d_out;               // [F][64]

  const int F        = in_sizes[1] / 3;   // 600000 (multiple of 64)
  const int numTiles = F >> 6;            // 9375
  const int G        = (numTiles < GRID_MAIN) ? numTiles : GRID_MAIN;

  // workspace: filt[192]->pad 256 | params[128]->pad 256 | partS[G*64] | partQ[G*64]
  float* wsf    = (float*)d_ws;
  float* filt   = wsf;
  float* params = wsf + 256;
  float* partS  = wsf + 512;
  float* partQ  = partS + (size_t)G * 64;

  v2f_filt_kernel<<<1, 64, 0, stream>>>(sw, filt);
  v2f_main_kernel<<<G, 256, 0, stream>>>(inputs, face, dw, bias, filt, out,
                                         partS, partQ, numTiles);
  v2f_stats_kernel<<<1, 64, 0, stream>>>(partS, partQ, gamma, beta, params, G,
                                         1.0f / (float)F);
  v2f_norm_kernel<<<2048, 256, 0, stream>>>(out, params, F * 16);
}